// DGLDigitCapsuleLayer_78185584657012
// MI455X (gfx1250) — compile-verified
//
#include <hip/hip_runtime.h>

// Capsule routing for B=256, I=1152, J=10, D=16, K=8 on gfx1250 (wave32).
// u_hat (189MB, ~fits the 192MB L2) materialized once via
// V_WMMA_F32_16X16X4_F32 (fp32 precision; problem is bandwidth-bound so no
// reason to drop to bf16). Routing iterations stream u_hat with fully
// coalesced 128-bit accesses in (I,J,B,D) layout. All reductions are
// deterministic fixed trees (no float atomics -> bitwise-stable output).

typedef __attribute__((ext_vector_type(2))) float v2f;
typedef __attribute__((ext_vector_type(8))) float v8f;

#define B_ 256
#define I_ 1152
#define J_ 10
#define D_ 16
#define K_ 8
#define ICHUNKS 16
#define ICHUNK (I_ / ICHUNKS)   // 72

// ---------- layout-prep kernels ----------

// x: (B, K, I) i-innermost  ->  xT: (I, K, B) b-innermost
__global__ __launch_bounds__(256) void k_transpose_x(const float* __restrict__ x,
                                                     float* __restrict__ xT) {
  int tid = blockIdx.x * blockDim.x + threadIdx.x;
  if (tid >= B_ * K_ * I_) return;
  int i = tid % I_;
  int k = (tid / I_) % K_;
  int b = tid / (I_ * K_);
  xT[(i * K_ + k) * B_ + b] = x[tid];
}

// W: (I, J, D, K) k-innermost -> WT: (I, J, K, D) d-innermost
__global__ __launch_bounds__(256) void k_transpose_w(const float* __restrict__ W,
                                                     float* __restrict__ WT) {
  int tid = blockIdx.x * blockDim.x + threadIdx.x;
  if (tid >= I_ * J_ * D_ * K_) return;
  int k = tid % K_;
  int d = (tid / K_) % D_;
  int p = tid / (D_ * K_);            // p = i*J + j
  WT[(p * K_ + k) * D_ + d] = W[tid];
}

__global__ __launch_bounds__(256) void k_zero(float* __restrict__ p, int n) {
  int tid = blockIdx.x * blockDim.x + threadIdx.x;
  if (tid < n) p[tid] = 0.0f;
}

// ---------- u_hat via fp32 WMMA ----------
// One wave per (i,j) pair. B operand (W-tile, 4x16) loaded once, reused
// across all 16 b-tiles. Two 16x16x4 WMMAs chain K=8.
// A layout (16x4 f32): lane&15 = M; half-wave selects K=0..1 vs K=2..3 in the
// two A registers. B layout (4x16 f32): lane&15 = N; same K split. C/D layout:
// VGPR r holds M=r (lanes 0-15) and M=r+8 (lanes 16-31), N=lane&15.
__global__ __launch_bounds__(256) void k_uhat_wmma(const float* __restrict__ xT,
                                                   const float* __restrict__ WT,
                                                   float* __restrict__ uhat) {
  int wave = (blockIdx.x * blockDim.x + threadIdx.x) >> 5;   // 0..11519 exact
  int lane = threadIdx.x & 31;
  int i = wave / J_;
  int j = wave - i * J_;
  int n = lane & 15;
  int half = lane >> 4;

  const float* wt = WT + (size_t)wave * (K_ * D_);
  v2f bm0, bm1;
  bm0.x = wt[(2 * half + 0) * D_ + n];
  bm0.y = wt[(2 * half + 1) * D_ + n];
  bm1.x = wt[(4 + 2 * half + 0) * D_ + n];
  bm1.y = wt[(4 + 2 * half + 1) * D_ + n];

  const float* xi = xT + (size_t)i * (K_ * B_);
  float* up = uhat + (size_t)wave * (B_ * D_);

  for (int t = 0; t < 16; ++t) {
    int brow = t * 16 + n;            // M index = lane&15
    v2f am0, am1;
    am0.x = xi[(2 * half + 0) * B_ + brow];
    am0.y = xi[(2 * half + 1) * B_ + brow];
    am1.x = xi[(4 + 2 * half + 0) * B_ + brow];
    am1.y = xi[(4 + 2 * half + 1) * B_ + brow];

    v8f acc = {};
    acc = __builtin_amdgcn_wmma_f32_16x16x4_f32(false, am0, false, bm0,
                                                (short)0, acc, false, false);
    acc = __builtin_amdgcn_wmma_f32_16x16x4_f32(false, am1, false, bm1,
                                                (short)0, acc, false, false);
#pragma unroll
    for (int r = 0; r < 8; ++r) {
      int b = t * 16 + r + 8 * half;
      up[b * D_ + n] = acc[r];
    }
  }
}

// ---------- routing iteration kernels ----------

// c[i,:] = softmax(b[i,:]) over J
__global__ __launch_bounds__(256) void k_softmax(const float* __restrict__ bb,
                                                 float* __restrict__ c) {
  int i = blockIdx.x * blockDim.x + threadIdx.x;
  if (i >= I_) return;
  float vals[J_];
  float m = -3.402823e38f;
#pragma unroll
  for (int j = 0; j < J_; ++j) { vals[j] = bb[i * J_ + j]; m = fmaxf(m, vals[j]); }
  float s = 0.0f;
#pragma unroll
  for (int j = 0; j < J_; ++j) { vals[j] = expf(vals[j] - m); s += vals[j]; }
  float inv = 1.0f / s;
#pragma unroll
  for (int j = 0; j < J_; ++j) c[i * J_ + j] = vals[j] * inv;
}

// Stage 1 of s[j,b,d] = sum_i c[i,j]*u_hat[i,j,b,d]: 16 deterministic
// i-chunks, 128-bit loads. grid (ICHUNKS, B/64, J); block 256 = 64 b x 4
// d-quads -> each i step reads 4KB contiguous per block.
__global__ __launch_bounds__(256) void k_s_partial(const float4* __restrict__ uhat4,
                                                   const float* __restrict__ c,
                                                   float4* __restrict__ sp4) {
  int chunk = blockIdx.x, bquad = blockIdx.y, j = blockIdx.z;
  int t = threadIdx.x;
  int dg = t & 3;                       // d-quad index (4 floats)
  int b = bquad * 64 + (t >> 2);
  float4 acc = make_float4(0.f, 0.f, 0.f, 0.f);
  int i0 = chunk * ICHUNK;
  for (int ii = 0; ii < ICHUNK; ++ii) {
    int p = (i0 + ii) * J_ + j;
    float cc = c[p];
    float4 u = uhat4[(size_t)p * (B_ * D_ / 4) + b * 4 + dg];
    acc.x += cc * u.x; acc.y += cc * u.y;
    acc.z += cc * u.z; acc.w += cc * u.w;
  }
  sp4[((size_t)(chunk * J_ + j) * B_ + b) * 4 + dg] = acc;
}

// Stage 2 + squash fused: sum 16 partial float4s, norm over D=16 via
// in-thread quad sum + 2-step shfl_xor (d-quad lives in lane bits 0..1).
__global__ __launch_bounds__(256) void k_squash(const float4* __restrict__ sp4,
                                                float4* __restrict__ v4) {
  int j = blockIdx.x, bquad = blockIdx.y;
  int t = threadIdx.x;
  int dg = t & 3;
  int b = bquad * 64 + (t >> 2);
  float4 s = make_float4(0.f, 0.f, 0.f, 0.f);
#pragma unroll
  for (int ch = 0; ch < ICHUNKS; ++ch) {
    float4 p = sp4[((size_t)(ch * J_ + j) * B_ + b) * 4 + dg];
    s.x += p.x; s.y += p.y; s.z += p.z; s.w += p.w;
  }
  float sq = s.x * s.x + s.y * s.y + s.z * s.z + s.w * s.w;
  sq += __shfl_xor(sq, 1);
  sq += __shfl_xor(sq, 2);              // full ||s||^2 over D for this b
  float scale = (sq / (1.0f + sq)) / sqrtf(sq);   // matches reference (no eps)
  float4 o;
  o.x = s.x * scale; o.y = s.y * scale;
  o.z = s.z * scale; o.w = s.w * scale;
  v4[((size_t)j * B_ + b) * 4 + dg] = o;
}

// b[i,j] += (1/B) * sum_{b,d} v[j,b,d] * u_hat[i,j,b,d].
// One block per (i,j): 16KB contiguous u_hat tile, b128 loads,
// deterministic tree reduce.
__global__ __launch_bounds__(256) void k_b_update(const float4* __restrict__ uhat4,
                                                  const float4* __restrict__ v4,
                                                  float* __restrict__ bb) {
  int ij = blockIdx.x;                  // i*J + j
  int j = ij % J_;
  const float4* up = uhat4 + (size_t)ij * (B_ * D_ / 4);
  const float4* vj = v4 + (size_t)j * (B_ * D_ / 4);
  int t = threadIdx.x;
  float acc = 0.0f;
#pragma unroll
  for (int e = t; e < B_ * D_ / 4; e += 256) {
    float4 u = up[e];
    float4 w = vj[e];
    acc += u.x * w.x + u.y * w.y + u.z * w.z + u.w * w.w;
  }
  acc += __shfl_xor(acc, 1);
  acc += __shfl_xor(acc, 2);
  acc += __shfl_xor(acc, 4);
  acc += __shfl_xor(acc, 8);
  acc += __shfl_xor(acc, 16);
  __shared__ float red[8];
  if ((t & 31) == 0) red[t >> 5] = acc;
  __syncthreads();
  if (t == 0) {
    float tot = 0.0f;
#pragma unroll
    for (int w = 0; w < 8; ++w) tot += red[w];
    bb[ij] += tot * (1.0f / (float)B_);
  }
}

// out (B,J,D,1) <- v (J,B,D), 128-bit granularity (D innermost in both).
__global__ __launch_bounds__(256) void k_write_out(const float4* __restrict__ v4,
                                                   float4* __restrict__ out4) {
  int q = blockIdx.x * blockDim.x + threadIdx.x;
  if (q >= B_ * J_ * D_ / 4) return;
  int dg = q & 3;
  int j = (q >> 2) % J_;
  int b = q / (J_ * 4);
  out4[q] = v4[((size_t)j * B_ + b) * 4 + dg];
}

// ---------- launch ----------

extern "C" void kernel_launch(void* const* d_in, const int* in_sizes, int n_in,
                              void* d_out, int out_size, void* d_ws, size_t ws_size,
                              hipStream_t stream) {
  const float* x = (const float*)d_in[0];   // (B, K, I)
  const float* W = (const float*)d_in[1];   // (I, J, D, K)

  float* ws = (float*)d_ws;
  float* xT   = ws;                                   // I*K*B      = 2,359,296
  float* WT   = xT + (size_t)I_ * K_ * B_;            // I*J*K*D    = 1,474,560
  float* uhat = WT + (size_t)I_ * J_ * K_ * D_;       // I*J*B*D    = 47,185,920
  float* bb   = uhat + (size_t)I_ * J_ * B_ * D_;     // I*J        = 11,520
  float* c    = bb + (size_t)I_ * J_;                 // I*J        = 11,520
  float* sp   = c + (size_t)I_ * J_;                  // 16*J*B*D   = 655,360
  float* v    = sp + (size_t)ICHUNKS * J_ * B_ * D_;  // J*B*D      = 40,960

  k_transpose_x<<<(B_ * K_ * I_ + 255) / 256, 256, 0, stream>>>(x, xT);
  k_transpose_w<<<(I_ * J_ * D_ * K_ + 255) / 256, 256, 0, stream>>>(W, WT);
  k_zero<<<(I_ * J_ + 255) / 256, 256, 0, stream>>>(bb, I_ * J_);

  // 11520 waves exactly -> 1440 blocks of 8 waves, EXEC all-ones for WMMA.
  k_uhat_wmma<<<(I_ * J_) / 8, 256, 0, stream>>>(xT, WT, uhat);

  for (int it = 0; it < 3; ++it) {
    k_softmax<<<(I_ + 255) / 256, 256, 0, stream>>>(bb, c);
    k_s_partial<<<dim3(ICHUNKS, B_ / 64, J_), 256, 0, stream>>>(
        (const float4*)uhat, c, (float4*)sp);
    k_squash<<<dim3(J_, B_ / 64), 256, 0, stream>>>((const float4*)sp,
                                                    (float4*)v);
    if (it < 2)  // final b-update is unused by the reference output
      k_b_update<<<I_ * J_, 256, 0, stream>>>((const float4*)uhat,
                                              (const float4*)v, bb);
  }

  k_write_out<<<(B_ * J_ * D_ / 4 + 255) / 256, 256, 0, stream>>>(
      (const float4*)v, (float4*)d_out);
}